// MT3Attention_59476707115027
// MI455X (gfx1250) — compile-verified
//
#include <hip/hip_runtime.h>
#include <hip/hip_bf16.h>

typedef __attribute__((ext_vector_type(2))) float v2f;
typedef __attribute__((ext_vector_type(8))) float v8f;

#define B_      8
#define S_      2048
#define D_      512
#define H_      8
#define DK_     48
#define INNER_  384
#define NREL    4096   // rel in [-2048, 2047]

static __device__ __forceinline__ v8f wmma_f32(v2f a, v2f b, v8f c) {
  // D = A(16x4,f32) * B(4x16,f32) + C(16x16,f32)
  return __builtin_amdgcn_wmma_f32_16x16x4_f32(
      /*neg_a=*/false, a, /*neg_b=*/false, b,
      /*c_mod=*/(short)0, c, /*reuse_a=*/false, /*reuse_b=*/false);
}

// ---------------------------------------------------------------------------
// Generic fp32 GEMM: C[M,N] = A[M,K] * B[K,N], all row-major.
// Block = 128 threads (4 waves). Block tile 64x64; each wave computes 16x64.
// LDS-staged K-chunks of 16 (4 WMMA K=4 steps per chunk).
// A tile: even-stride pad (18) -> conflict-free column reads + b64 pair loads.
// B tile: pair-interleaved Bs2[k>>1][n][k&1] -> one ds_load_b64 per fragment.
// ---------------------------------------------------------------------------
__global__ __launch_bounds__(128)
void gemm_f32_kernel(const float* __restrict__ A, const float* __restrict__ Bm,
                     float* __restrict__ C, int M, int N, int K,
                     int lda, int ldb, int ldc) {
  __shared__ float As[64][18];        // stride 18: even (8B pairs) + no conflicts
  __shared__ float Bs2[8][64][2];     // [kpair][n][k&1]
  const int tid   = threadIdx.x;
  const int lane  = tid & 31;
  const int wid   = tid >> 5;
  const int hpart = lane >> 4;        // 0..1 (wave halves)
  const int l16   = lane & 15;
  const int bm = blockIdx.x * 64;
  const int bn = blockIdx.y * 64;
  const int wm = wid << 4;

  v8f acc[4] = {};                    // 4 n-tiles of 16x16

  const int nChunks = K >> 4;
  for (int ch = 0; ch < nChunks; ++ch) {
    const int k0 = ch << 4;
    // stage A chunk 64x16 (256 float4, coalesced)
    for (int i = tid; i < 256; i += 128) {
      int row = i >> 2, c4 = (i & 3) << 2;
      const float* ap = A + (size_t)(bm + row) * lda + k0 + c4;
      float4 v = *(const float4*)ap;
      As[row][c4 + 0] = v.x; As[row][c4 + 1] = v.y;
      As[row][c4 + 2] = v.z; As[row][c4 + 3] = v.w;
      if (ch + 1 < nChunks) __builtin_prefetch(ap + 16, 0, 1);
    }
    // stage B chunk 16x64 (256 float4) into pair-interleaved layout
    for (int i = tid; i < 256; i += 128) {
      int row = i >> 4, c4 = (i & 15) << 2;
      const float* bp = Bm + (size_t)(k0 + row) * ldb + bn + c4;
      float4 v = *(const float4*)bp;
      int kp = row >> 1, klo = row & 1;
      Bs2[kp][c4 + 0][klo] = v.x; Bs2[kp][c4 + 1][klo] = v.y;
      Bs2[kp][c4 + 2][klo] = v.z; Bs2[kp][c4 + 3][klo] = v.w;
      if (ch + 1 < nChunks) __builtin_prefetch(bp + 16 * ldb, 0, 1);
    }
    __syncthreads();

    for (int kk = 0; kk < 4; ++kk) {
      const int kb = (kk << 2) + hpart * 2;   // even
      v2f a = *(const v2f*)&As[wm + l16][kb];
      for (int nt = 0; nt < 4; ++nt) {
        v2f b = *(const v2f*)&Bs2[kb >> 1][(nt << 4) + l16][0];
        acc[nt] = wmma_f32(a, b, acc[nt]);
      }
    }
    __syncthreads();
  }

  for (int nt = 0; nt < 4; ++nt) {
    for (int r = 0; r < 8; ++r) {
      int row = bm + wm + r + (hpart << 3);
      int col = bn + (nt << 4) + l16;
      C[(size_t)row * ldc + col] = acc[nt][r];
    }
  }
}

// ---------------------------------------------------------------------------
// Precompute T5 relative-position bias table: tab[h][rel + 2048]
// ---------------------------------------------------------------------------
__global__ __launch_bounds__(256)
void bias_table_kernel(const float* __restrict__ rel_bias, float* __restrict__ tab) {
  int i = blockIdx.x * 256 + threadIdx.x;
  if (i >= H_ * NREL) return;
  int h   = i >> 12;
  int rel = (i & (NREL - 1)) - (NREL / 2);
  int rb  = (rel > 0) ? 16 : 0;             // num_buckets//2
  int a   = (rel < 0) ? -rel : rel;
  int idx;
  if (a < 8) {                              // max_exact
    idx = a;
  } else {
    int v = 8 + (int)(__logf((float)a * 0.125f) * (8.0f / __logf(16.0f)));
    idx = (v < 15) ? v : 15;
  }
  tab[i] = rel_bias[(rb + idx) * H_ + h];
}

// ---------------------------------------------------------------------------
// Flash attention per (b,h). Block = 128 threads (4 waves), covers 64 q rows;
// each wave owns a 16-row q tile. K/V 16x48 tiles staged in LDS (shared by
// all 4 waves). Online softmax with cross-lane reductions in 16-lane halves.
// Q/K/V/O layout: [B*S, 384] with head h at column h*48.
// Ks: even-stride pad (50). Vs: pair-interleaved. Ps: even-stride pad (18).
// ---------------------------------------------------------------------------
__global__ __launch_bounds__(128)
void attn_kernel(const float* __restrict__ Q, const float* __restrict__ Km,
                 const float* __restrict__ V, const float* __restrict__ tab,
                 float* __restrict__ O) {
  __shared__ float Ks[16][50];        // [kcol][d], stride 50: even + no conflicts
  __shared__ float Vs2[8][48][2];     // [k>>1][d][k&1]
  __shared__ float Ps[4][16][18];     // per-wave P tile, stride 18
  const int tid   = threadIdx.x;
  const int lane  = tid & 31;
  const int wid   = tid >> 5;
  const int hpart = lane >> 4;
  const int l16   = lane & 15;
  const int bq = blockIdx.x * 64;           // q tile base within sequence
  const int b  = blockIdx.y >> 3;
  const int h  = blockIdx.y & 7;
  const int q0 = bq + (wid << 4);           // this wave's 16 q rows

  const size_t rowBase = (size_t)b * S_ * INNER_;
  const int    hoff    = h * DK_;
  const float* tabh    = tab + h * NREL + (NREL / 2);

  // preload Q fragments: 12 K-steps of 4 over d_kv=48 (v2f: d is even -> 8B)
  v2f qa[12];
  {
    const float* qrow = Q + rowBase + (size_t)(q0 + l16) * INNER_ + hoff;
    for (int ks = 0; ks < 12; ++ks) {
      int d = (ks << 2) + hpart * 2;
      qa[ks] = *(const v2f*)(qrow + d);
    }
  }

  v8f   o[3] = {};
  float mrun[8], lrun[8];
  for (int r = 0; r < 8; ++r) { mrun[r] = -3.0e38f; lrun[r] = 0.0f; }

  for (int kt = 0; kt < S_ / 16; ++kt) {
    __syncthreads();   // previous iteration's Vs/Ps consumers are done
    // stage K and V 16x48 tiles (float4, coalesced)
    {
      const float* kg = Km + rowBase + (size_t)(kt * 16) * INNER_ + hoff;
      const float* vg = V  + rowBase + (size_t)(kt * 16) * INNER_ + hoff;
      for (int i = tid; i < 192; i += 128) {
        int row = i / 12, c4 = (i % 12) << 2;
        const float* kp = kg + (size_t)row * INNER_ + c4;
        const float* vp = vg + (size_t)row * INNER_ + c4;
        float4 kv = *(const float4*)kp;
        float4 vv = *(const float4*)vp;
        Ks[row][c4 + 0] = kv.x; Ks[row][c4 + 1] = kv.y;
        Ks[row][c4 + 2] = kv.z; Ks[row][c4 + 3] = kv.w;
        int kpr = row >> 1, klo = row & 1;
        Vs2[kpr][c4 + 0][klo] = vv.x; Vs2[kpr][c4 + 1][klo] = vv.y;
        Vs2[kpr][c4 + 2][klo] = vv.z; Vs2[kpr][c4 + 3][klo] = vv.w;
        if (kt + 1 < S_ / 16) {
          __builtin_prefetch(kp + 16 * INNER_, 0, 1);
          __builtin_prefetch(vp + 16 * INNER_, 0, 1);
        }
      }
    }
    __syncthreads();

    // S tile = Q * K^T   (B[d][n] = K[n][d]; (kb,kb+1) contiguous in Ks row)
    v8f s = {};
    for (int ks = 0; ks < 12; ++ks) {
      int kb = (ks << 2) + hpart * 2;
      v2f bfrag = *(const v2f*)&Ks[l16][kb];
      s = wmma_f32(qa[ks], bfrag, s);
    }

    // bias + online softmax (row r+8*hpart lives in element r across 16 lanes)
    for (int r = 0; r < 8; ++r) {
      int   qrow = q0 + r + (hpart << 3);
      int   kcol = (kt << 4) + l16;
      float val  = s[r] + tabh[kcol - qrow];
      float mx = val;
      mx = fmaxf(mx, __shfl_xor(mx, 1, 32));
      mx = fmaxf(mx, __shfl_xor(mx, 2, 32));
      mx = fmaxf(mx, __shfl_xor(mx, 4, 32));
      mx = fmaxf(mx, __shfl_xor(mx, 8, 32));
      float mn  = fmaxf(mrun[r], mx);
      float pe  = __expf(val - mn);
      float sum = pe;
      sum += __shfl_xor(sum, 1, 32);
      sum += __shfl_xor(sum, 2, 32);
      sum += __shfl_xor(sum, 4, 32);
      sum += __shfl_xor(sum, 8, 32);
      float sc = __expf(mrun[r] - mn);
      lrun[r] = lrun[r] * sc + sum;
      mrun[r] = mn;
      o[0][r] *= sc; o[1][r] *= sc; o[2][r] *= sc;
      Ps[wid][r + (hpart << 3)][l16] = pe;   // C-layout -> row-major LDS
    }
    __syncthreads();

    // O += P * V
    for (int ks2 = 0; ks2 < 4; ++ks2) {
      int kb = (ks2 << 2) + hpart * 2;
      v2f a = *(const v2f*)&Ps[wid][l16][kb];
      for (int dt = 0; dt < 3; ++dt) {
        v2f bfrag = *(const v2f*)&Vs2[kb >> 1][(dt << 4) + l16][0];
        o[dt] = wmma_f32(a, bfrag, o[dt]);
      }
    }
  }

  // normalize and store (heads concatenate for free in [B*S, 384] layout)
  for (int dt = 0; dt < 3; ++dt) {
    for (int r = 0; r < 8; ++r) {
      int row = q0 + r + (hpart << 3);
      O[rowBase + (size_t)row * INNER_ + hoff + (dt << 4) + l16] = o[dt][r] / lrun[r];
    }
  }
}

// ---------------------------------------------------------------------------
extern "C" void kernel_launch(void* const* d_in, const int* in_sizes, int n_in,
                              void* d_out, int out_size, void* d_ws, size_t ws_size,
                              hipStream_t stream) {
  const float* X  = (const float*)d_in[0];   // [8,2048,512]
  const float* wq = (const float*)d_in[1];   // [512,384]
  const float* wk = (const float*)d_in[2];
  const float* wv = (const float*)d_in[3];
  const float* wo = (const float*)d_in[4];   // [384,512]
  const float* rb = (const float*)d_in[5];   // [32,8]
  float* out = (float*)d_out;                // [8,2048,512]

  float* ws = (float*)d_ws;
  const size_t qkvElems = (size_t)B_ * S_ * INNER_;   // 6,291,456 (24 MB)
  float* Qb  = ws;
  float* Kb  = Qb + qkvElems;
  float* Vb  = Kb + qkvElems;
  float* Ob  = Vb + qkvElems;
  float* tab = Ob + qkvElems;                         // + 128 KB

  dim3 blk(128);
  dim3 gQKV(B_ * S_ / 64, INNER_ / 64);               // 256 x 6
  gemm_f32_kernel<<<gQKV, blk, 0, stream>>>(X, wq, Qb, B_ * S_, INNER_, D_, D_, INNER_, INNER_);
  gemm_f32_kernel<<<gQKV, blk, 0, stream>>>(X, wk, Kb, B_ * S_, INNER_, D_, D_, INNER_, INNER_);
  gemm_f32_kernel<<<gQKV, blk, 0, stream>>>(X, wv, Vb, B_ * S_, INNER_, D_, D_, INNER_, INNER_);

  bias_table_kernel<<<(H_ * NREL + 255) / 256, 256, 0, stream>>>(rb, tab);

  attn_kernel<<<dim3(S_ / 64, B_ * H_), blk, 0, stream>>>(Qb, Kb, Vb, tab, Ob);

  dim3 gO(B_ * S_ / 64, D_ / 64);                     // 256 x 8
  gemm_f32_kernel<<<gO, blk, 0, stream>>>(Ob, wo, out, B_ * S_, D_, INNER_, INNER_, D_, D_);
}